// GNNSceneEncoder_72851235275075
// MI455X (gfx1250) — compile-verified
//
#include <hip/hip_runtime.h>
#include <math.h>

// ---------------------------------------------------------------------------
// GNN Scene Encoder for MI455X (gfx1250, wave32)
//   - dense GEMMs via V_WMMA_F32_16X16X4_F32 (exact f32 matrix pipe)
//     all shapes divide exactly -> branch-free inner loop
//   - edge-attention folded:  a_e = ea2 @ (lin_e . att_e)  [32x4]
//   - segment softmax via order-preserving uint atomicMax + f32 atomicAdd
// ---------------------------------------------------------------------------

typedef __attribute__((ext_vector_type(2))) float v2f;
typedef __attribute__((ext_vector_type(8))) float v8f;

#define F_IN 518
#define HIDN 256
#define DD   128
#define HH   4
#define HD   512   // HH*DD
#define ED   32

// ---------------- WMMA f32 GEMM:  C[M,Ncol] = A[M,K] @ B[K,Ncol] (+bias)(+relu)
// wave tile: 16(M) x 64(N) with 4 accumulators; block = 4 waves -> 32 x 128
// REQUIRES: M % 32 == 0, Ncol % 128 == 0 (grid covers exactly; no bounds checks)
__global__ __launch_bounds__(128)
void gemm_wmma_f32(const float* __restrict__ A, const float* __restrict__ B,
                   const float* __restrict__ bias, float* __restrict__ C,
                   int M, int Ncol, int K, int relu) {
    const int lane = threadIdx.x & 31;
    const int wave = threadIdx.x >> 5;
    const int l16  = lane & 15;
    const int half = lane >> 4;                 // 0/1
    const int rowBase = blockIdx.x * 32 + (wave >> 1) * 16;
    const int colBase = blockIdx.y * 128 + (wave & 1) * 64;

    v8f acc[4];
    #pragma unroll
    for (int t = 0; t < 4; ++t) {
        v8f z = {0.f,0.f,0.f,0.f,0.f,0.f,0.f,0.f};
        acc[t] = z;
    }

    const int m = rowBase + l16;                // A row this lane feeds (always valid)
    // A fragment: lane half=0 holds K = k0,k0+1 ; half=1 holds K = k0+2,k0+3
    const float* Ap = A + (size_t)m * K + 2 * half;          // 8B-aligned (K even)
    // B fragment: VGPR v holds row K = k0+2*half+v, col = colBase + t*16 + l16
    const float* Bp = B + (size_t)(2 * half) * Ncol + colBase + l16;
    const size_t bStep = 4 * (size_t)Ncol;

    const int kFull = K & ~3;
    #pragma unroll 2
    for (int k0 = 0; k0 < kFull; k0 += 4) {
        v2f a = *(const v2f*)Ap;
        v2f b[4];
        #pragma unroll
        for (int t = 0; t < 4; ++t) {
            b[t].x = Bp[t * 16];
            b[t].y = Bp[(size_t)Ncol + t * 16];
        }
        #pragma unroll
        for (int t = 0; t < 4; ++t)
            acc[t] = __builtin_amdgcn_wmma_f32_16x16x4_f32(false, a, false, b[t],
                                                           (short)0, acc[t], false, false);
        Ap += 4;
        Bp += bStep;
    }
    if (kFull < K) {                            // K tail (K=518 -> 2 leftover)
        const int ka  = kFull + 2 * half;
        const int kc0 = (ka     < K) ? ka     : (K - 1);     // clamped, in-bounds
        const int kc1 = (ka + 1 < K) ? ka + 1 : (K - 1);
        const float* At = A + (size_t)m * K;
        float ax = At[kc0];
        float ay = At[kc1];
        v2f a;                                   // zero A where k OOB (selects, not branches)
        a.x = (ka     < K) ? ax : 0.f;
        a.y = (ka + 1 < K) ? ay : 0.f;
        // B rows clamped: 0 * finite = 0, no masking needed
        const float* Bt0 = B + (size_t)kc0 * Ncol + colBase + l16;
        const float* Bt1 = B + (size_t)kc1 * Ncol + colBase + l16;
        v2f b[4];
        #pragma unroll
        for (int t = 0; t < 4; ++t) { b[t].x = Bt0[t * 16]; b[t].y = Bt1[t * 16]; }
        #pragma unroll
        for (int t = 0; t < 4; ++t)
            acc[t] = __builtin_amdgcn_wmma_f32_16x16x4_f32(false, a, false, b[t],
                                                           (short)0, acc[t], false, false);
    }
    // C/D layout: VGPR v, lanes 0-15 -> M=rowBase+v ; lanes 16-31 -> M=rowBase+v+8 ; N=l16
    #pragma unroll
    for (int t = 0; t < 4; ++t) {
        const int n = colBase + t * 16 + l16;
        const float bv = bias ? bias[n] : 0.f;
        float* Cp = C + (size_t)(rowBase + 8 * half) * Ncol + n;
        #pragma unroll
        for (int v = 0; v < 8; ++v) {
            float val = acc[t][v] + bv;
            if (relu) val = fmaxf(val, 0.f);
            Cp[(size_t)v * Ncol] = val;
        }
    }
}

// ---------------- order-preserving float<->uint key for atomicMax
__device__ __forceinline__ unsigned fkey(float x) {
    unsigned u = __float_as_uint(x);
    return (u >> 31) ? ~u : (u | 0x80000000u);
}
__device__ __forceinline__ float funkey(unsigned k) {
    unsigned u = (k >> 31) ? (k & 0x7fffffffu) : ~k;
    return __uint_as_float(u);
}

// ---------------- degree count + self-loop edge-attr mean accumulation
__global__ void k_deg_loopea(const int* __restrict__ dst, const int* __restrict__ et,
                             const float* __restrict__ rel, float* __restrict__ cnt,
                             float* __restrict__ loop_ea, int E) {
    int tid = blockIdx.x * blockDim.x + threadIdx.x;
    int e = tid >> 5, k = tid & 31;
    if (e >= E) return;
    int d = dst[e];
    atomicAdd(&loop_ea[d * ED + k], rel[et[e] * ED + k]);
    if (k == 0) atomicAdd(&cnt[d], 1.0f);
}

__global__ void k_loopea_div(float* __restrict__ loop_ea, const float* __restrict__ cnt, int Nn) {
    int tid = blockIdx.x * blockDim.x + threadIdx.x;
    int n = tid >> 5, k = tid & 31;
    if (n >= Nn) return;
    loop_ea[n * ED + k] /= fmaxf(cnt[n], 1.0f);
}

// ---------------- fold M_e[k][h] = sum_d lin_e[k, h*D+d] * att_e[h,d]
__global__ void k_fold_me(const float* __restrict__ line, const float* __restrict__ atte,
                          float* __restrict__ Me) {
    int t = threadIdx.x;          // 0..127
    int k = t >> 2, h = t & 3;
    float s = 0.f;
    for (int d = 0; d < DD; ++d) s += line[k * HD + h * DD + d] * atte[h * DD + d];
    Me[k * 4 + h] = s;
}

// ---------------- a_e for real edges (gathers rel_emb row on the fly)
__global__ void k_ae_edges(const int* __restrict__ et, const float* __restrict__ rel,
                           const float* __restrict__ Me, float* __restrict__ ae, int E) {
    int tid = blockIdx.x * blockDim.x + threadIdx.x;
    int e = tid >> 2, h = tid & 3;
    if (e >= E) return;
    const float* r = rel + et[e] * ED;
    float s = 0.f;
    #pragma unroll
    for (int k = 0; k < ED; ++k) s += r[k] * Me[k * 4 + h];
    ae[e * 4 + h] = s;
}

__global__ void k_ae_loop(const float* __restrict__ loop_ea, const float* __restrict__ Me,
                          float* __restrict__ ael, int Nn) {
    int tid = blockIdx.x * blockDim.x + threadIdx.x;
    int n = tid >> 2, h = tid & 3;
    if (n >= Nn) return;
    const float* r = loop_ea + n * ED;
    float s = 0.f;
    #pragma unroll
    for (int k = 0; k < ED; ++k) s += r[k] * Me[k * 4 + h];
    ael[n * 4 + h] = s;
}

// ---------------- a_s / a_d: one wave per node, shuffle reduction
__global__ __launch_bounds__(256)
void k_as_ad(const float* __restrict__ hh, const float* __restrict__ atts,
             const float* __restrict__ attd, float* __restrict__ a_s,
             float* __restrict__ a_d, int Nn) {
    int lane = threadIdx.x & 31;
    int wv   = threadIdx.x >> 5;
    int n = blockIdx.x * 8 + wv;
    if (n >= Nn) return;
    float ss[4] = {0.f,0.f,0.f,0.f}, sd[4] = {0.f,0.f,0.f,0.f};
    #pragma unroll
    for (int h = 0; h < HH; ++h) {
        #pragma unroll
        for (int j = 0; j < 4; ++j) {
            int d = lane + j * 32;
            float hv = hh[(size_t)n * HD + h * DD + d];
            ss[h] += hv * atts[h * DD + d];
            sd[h] += hv * attd[h * DD + d];
        }
    }
    #pragma unroll
    for (int off = 16; off > 0; off >>= 1) {
        #pragma unroll
        for (int h = 0; h < HH; ++h) {
            ss[h] += __shfl_down(ss[h], off, 32);
            sd[h] += __shfl_down(sd[h], off, 32);
        }
    }
    if (lane == 0) {
        #pragma unroll
        for (int h = 0; h < HH; ++h) { a_s[n * 4 + h] = ss[h]; a_d[n * 4 + h] = sd[h]; }
    }
}

// ---------------- alpha = leaky_relu(a_s[s]+a_d[d]+a_e), running segment max
__global__ void k_alpha_edges(const int* __restrict__ src, const int* __restrict__ dst,
                              const float* __restrict__ as_, const float* __restrict__ ad_,
                              const float* __restrict__ ae, float* __restrict__ alpha,
                              unsigned* __restrict__ amax, int E) {
    int tid = blockIdx.x * blockDim.x + threadIdx.x;
    int e = tid >> 2, h = tid & 3;
    if (e >= E) return;
    int d = dst[e];
    float v = as_[src[e] * 4 + h] + ad_[d * 4 + h] + ae[e * 4 + h];
    v = v > 0.f ? v : 0.2f * v;
    alpha[e * 4 + h] = v;
    atomicMax(&amax[d * 4 + h], fkey(v));
}

__global__ void k_alpha_loop(const float* __restrict__ as_, const float* __restrict__ ad_,
                             const float* __restrict__ ael, float* __restrict__ alphal,
                             unsigned* __restrict__ amax, int Nn) {
    int tid = blockIdx.x * blockDim.x + threadIdx.x;
    int n = tid >> 2, h = tid & 3;
    if (n >= Nn) return;
    float v = as_[n * 4 + h] + ad_[n * 4 + h] + ael[n * 4 + h];
    v = v > 0.f ? v : 0.2f * v;
    alphal[n * 4 + h] = v;
    atomicMax(&amax[n * 4 + h], fkey(v));
}

// ---------------- p = exp(alpha - amax[d]) (in place), denom accumulation
__global__ void k_exp_edges(const int* __restrict__ dst, float* __restrict__ alpha,
                            const unsigned* __restrict__ amax, float* __restrict__ denom, int E) {
    int tid = blockIdx.x * blockDim.x + threadIdx.x;
    int e = tid >> 2, h = tid & 3;
    if (e >= E) return;
    int d = dst[e];
    float p = expf(alpha[e * 4 + h] - funkey(amax[d * 4 + h]));
    alpha[e * 4 + h] = p;
    atomicAdd(&denom[d * 4 + h], p);
}

__global__ void k_exp_loop(float* __restrict__ alphal, const unsigned* __restrict__ amax,
                           float* __restrict__ denom, int Nn) {
    int tid = blockIdx.x * blockDim.x + threadIdx.x;
    int n = tid >> 2, h = tid & 3;
    if (n >= Nn) return;
    float p = expf(alphal[n * 4 + h] - funkey(amax[n * 4 + h]));
    alphal[n * 4 + h] = p;
    atomicAdd(&denom[n * 4 + h], p);
}

// ---------------- weighted scatter of messages: acc[d] += hh[s] * alpha
__global__ __launch_bounds__(256)
void k_scatter_edges(const int* __restrict__ src, const int* __restrict__ dst,
                     const float* __restrict__ p, const float* __restrict__ denom,
                     const float* __restrict__ hh, float* __restrict__ acc, int E) {
    int e = blockIdx.x * 2 + (threadIdx.x >> 7);
    int t = threadIdx.x & 127;
    if (e >= E) return;
    int s = src[e], d = dst[e];
    #pragma unroll
    for (int h = 0; h < HH; ++h) {
        float w = p[e * 4 + h] / (denom[d * 4 + h] + 1e-16f);
        atomicAdd(&acc[(size_t)d * HD + h * DD + t], hh[(size_t)s * HD + h * DD + t] * w);
    }
}

__global__ __launch_bounds__(256)
void k_scatter_loop(const float* __restrict__ p, const float* __restrict__ denom,
                    const float* __restrict__ hh, float* __restrict__ acc, int Nn) {
    int n = blockIdx.x * 2 + (threadIdx.x >> 7);
    int t = threadIdx.x & 127;
    if (n >= Nn) return;
    #pragma unroll
    for (int h = 0; h < HH; ++h) {
        float w = p[n * 4 + h] / (denom[n * 4 + h] + 1e-16f);
        atomicAdd(&acc[(size_t)n * HD + h * DD + t], hh[(size_t)n * HD + h * DD + t] * w);
    }
}

// ---------------- head-mean + bias + relu
__global__ void k_finalize(const float* __restrict__ acc, const float* __restrict__ bias,
                           float* __restrict__ out, int Nn) {
    int tid = blockIdx.x * blockDim.x + threadIdx.x;
    int n = tid >> 7, t = tid & 127;
    if (n >= Nn) return;
    const float* a = acc + (size_t)n * HD + t;
    float m = 0.25f * (a[0] + a[128] + a[256] + a[384]) + bias[t];
    out[n * DD + t] = fmaxf(m, 0.f);
}

// ---------------------------------------------------------------------------
extern "C" void kernel_launch(void* const* d_in, const int* in_sizes, int n_in,
                              void* d_out, int out_size, void* d_ws, size_t ws_size,
                              hipStream_t stream) {
    const int Nn = in_sizes[0] / F_IN;     // 20000
    const int E  = in_sizes[2];            // 320000

    const float* x     = (const float*)d_in[0];
    const int*   ei    = (const int*)d_in[1];
    const int*   etype = (const int*)d_in[2];
    const float* w1    = (const float*)d_in[3];
    const float* b1    = (const float*)d_in[4];
    const float* w2    = (const float*)d_in[5];
    const float* b2    = (const float*)d_in[6];
    const float* rel   = (const float*)d_in[7];
    const int* src = ei;
    const int* dst = ei + E;

    // ---- workspace layout (bump allocator over d_ws) ----
    float* W = (float*)d_ws;
    size_t o = 0;
    auto alloc = [&](size_t nf) { float* p = W + o; o += nf; return p; };
    float* h1      = alloc((size_t)Nn * HIDN);   // encoder hidden
    float* h0      = alloc((size_t)Nn * DD);     // encoder out
    float* hmid    = alloc((size_t)Nn * DD);     // layer1 out
    float* hh      = alloc((size_t)Nn * HD);     // per-layer h@lin
    float* acc     = alloc((size_t)Nn * HD);     // message accumulator
    float* cnt     = alloc((size_t)Nn);
    float* loop_ea = alloc((size_t)Nn * ED);
    float* Me      = alloc(ED * HH);
    float* a_e     = alloc((size_t)E * HH);
    float* a_el    = alloc((size_t)Nn * HH);
    float* a_s     = alloc((size_t)Nn * HH);
    float* a_d     = alloc((size_t)Nn * HH);
    float* alpha_e = alloc((size_t)E * HH);      // reused in place as p
    float* alpha_l = alloc((size_t)Nn * HH);
    float* denom   = alloc((size_t)Nn * HH);
    unsigned* amaxkey = (unsigned*)alloc((size_t)Nn * HH);

    // ---- node encoder: h0 = relu(x@w1+b1)@w2 + b2 ----
    {
        dim3 g1(Nn / 32, HIDN / 128);
        gemm_wmma_f32<<<g1, 128, 0, stream>>>(x, w1, b1, h1, Nn, HIDN, F_IN, 1);
        dim3 g2(Nn / 32, DD / 128);
        gemm_wmma_f32<<<g2, 128, 0, stream>>>(h1, w2, b2, h0, Nn, DD, HIDN, 0);
    }

    // ---- self-loop edge attributes (shared by both layers) ----
    hipMemsetAsync(cnt, 0, (size_t)Nn * sizeof(float), stream);
    hipMemsetAsync(loop_ea, 0, (size_t)Nn * ED * sizeof(float), stream);
    k_deg_loopea<<<((size_t)E * 32 + 255) / 256, 256, 0, stream>>>(dst, etype, rel, cnt, loop_ea, E);
    k_loopea_div<<<((size_t)Nn * 32 + 255) / 256, 256, 0, stream>>>(loop_ea, cnt, Nn);

    // ---- one GAT layer ----
    auto run_layer = [&](const float* hcur, const float* lin, const float* line,
                         const float* atts, const float* attd, const float* atte,
                         const float* bias, float* hout) {
        k_fold_me<<<1, 128, 0, stream>>>(line, atte, Me);
        k_ae_edges<<<((size_t)E * 4 + 255) / 256, 256, 0, stream>>>(etype, rel, Me, a_e, E);
        k_ae_loop<<<((size_t)Nn * 4 + 255) / 256, 256, 0, stream>>>(loop_ea, Me, a_el, Nn);

        dim3 gg(Nn / 32, HD / 128);
        gemm_wmma_f32<<<gg, 128, 0, stream>>>(hcur, lin, nullptr, hh, Nn, HD, DD, 0);

        k_as_ad<<<(Nn + 7) / 8, 256, 0, stream>>>(hh, atts, attd, a_s, a_d, Nn);

        hipMemsetAsync(amaxkey, 0, (size_t)Nn * HH * sizeof(unsigned), stream);
        hipMemsetAsync(denom, 0, (size_t)Nn * HH * sizeof(float), stream);
        hipMemsetAsync(acc, 0, (size_t)Nn * HD * sizeof(float), stream);

        k_alpha_edges<<<((size_t)E * 4 + 255) / 256, 256, 0, stream>>>(src, dst, a_s, a_d, a_e, alpha_e, amaxkey, E);
        k_alpha_loop<<<((size_t)Nn * 4 + 255) / 256, 256, 0, stream>>>(a_s, a_d, a_el, alpha_l, amaxkey, Nn);
        k_exp_edges<<<((size_t)E * 4 + 255) / 256, 256, 0, stream>>>(dst, alpha_e, amaxkey, denom, E);
        k_exp_loop<<<((size_t)Nn * 4 + 255) / 256, 256, 0, stream>>>(alpha_l, amaxkey, denom, Nn);
        k_scatter_edges<<<(E + 1) / 2, 256, 0, stream>>>(src, dst, alpha_e, denom, hh, acc, E);
        k_scatter_loop<<<(Nn + 1) / 2, 256, 0, stream>>>(alpha_l, denom, hh, acc, Nn);
        k_finalize<<<((size_t)Nn * DD + 255) / 256, 256, 0, stream>>>(acc, bias, hout, Nn);
    };

    run_layer(h0,   (const float*)d_in[8],  (const float*)d_in[9],
                    (const float*)d_in[10], (const float*)d_in[11],
                    (const float*)d_in[12], (const float*)d_in[13], hmid);
    run_layer(hmid, (const float*)d_in[14], (const float*)d_in[15],
                    (const float*)d_in[16], (const float*)d_in[17],
                    (const float*)d_in[18], (const float*)d_in[19], (float*)d_out);
}